// QuantumBranch_40295383171565
// MI455X (gfx1250) — compile-verified
//
#include <hip/hip_runtime.h>

typedef __attribute__((ext_vector_type(2))) float v2f;
typedef __attribute__((ext_vector_type(8))) float v8f;

#define TILES_PER_WAVE 4

__device__ __forceinline__ v2f mk2(float a, float b) { v2f r; r.x = a; r.y = b; return r; }

// ---------------------------------------------------------------------------
// Setup: build v0 = A1|000>, and M2/M3 = 16x16 real embeddings of A2/A3 with
// K-permuted columns so B-fragments need no cross-lane movement.
// K order: chunk k = [Re 2k, Re 2k+1, Im 8+2k, Im 8+2k+1].
// ws layout (floats): [0..7]=Re(v0), [8..15]=Im(v0), [16..271]=M2, [272..527]=M3
// ---------------------------------------------------------------------------
__global__ void qb_setup(const float* __restrict__ theta, float* __restrict__ ws) {
  const int tid = threadIdx.x;
  const int layer = tid >> 3;
  const int col = tid & 7;
  if (layer >= 3) return;

  float sr[8], si[8];
#pragma unroll
  for (int n = 0; n < 8; ++n) { sr[n] = (n == col) ? 1.0f : 0.0f; si[n] = 0.0f; }

  const float* th = theta + layer * 9;
#pragma unroll
  for (int w = 0; w < 3; ++w) {
    const int m = 4 >> w;  // bit value of wire w (wire0 = MSB)
    // RZ(a): diag(e^{-ia/2}, e^{+ia/2})
    {
      const float c = cosf(0.5f * th[w * 3 + 0]);
      const float s = sinf(0.5f * th[w * 3 + 0]);
#pragma unroll
      for (int n = 0; n < 8; ++n) {
        const float sg = (n & m) ? s : -s;
        const float r = sr[n], q = si[n];
        sr[n] = c * r - sg * q;
        si[n] = c * q + sg * r;
      }
    }
    // RX(a): [[c, -i s],[-i s, c]]
    {
      const float c = cosf(0.5f * th[w * 3 + 1]);
      const float s = sinf(0.5f * th[w * 3 + 1]);
#pragma unroll
      for (int n = 0; n < 8; ++n) {
        if (n & m) continue;
        const int q = n | m;
        const float r0 = sr[n], i0 = si[n], r1 = sr[q], i1 = si[q];
        sr[n] = c * r0 + s * i1;
        si[n] = c * i0 - s * r1;
        sr[q] = c * r1 + s * i0;
        si[q] = c * i1 - s * r0;
      }
    }
    // RZ(a)
    {
      const float c = cosf(0.5f * th[w * 3 + 2]);
      const float s = sinf(0.5f * th[w * 3 + 2]);
#pragma unroll
      for (int n = 0; n < 8; ++n) {
        const float sg = (n & m) ? s : -s;
        const float r = sr[n], q = si[n];
        sr[n] = c * r - sg * q;
        si[n] = c * q + sg * r;
      }
    }
  }
  // CNOT ring: (0,1),(1,2),(2,0); wire bits: w0=4, w1=2, w2=1
#define QB_SWAP(a, b)                             \
  { float tr = sr[a]; sr[a] = sr[b]; sr[b] = tr;  \
    float ti = si[a]; si[a] = si[b]; si[b] = ti; }
  QB_SWAP(4, 6) QB_SWAP(5, 7)  // CNOT(0,1)
  QB_SWAP(2, 3) QB_SWAP(6, 7)  // CNOT(1,2)
  QB_SWAP(1, 5) QB_SWAP(3, 7)  // CNOT(2,0)
#undef QB_SWAP

  if (layer == 0) {
    if (col == 0) {
#pragma unroll
      for (int r = 0; r < 8; ++r) { ws[r] = sr[r]; ws[8 + r] = si[r]; }
    }
  } else {
    float* M = ws + 16 + (layer - 1) * 256;  // row-major 16x16, K-permuted columns
    // old Re-column col -> new position nc1; old Im-column (col+8) -> nc1+2
    const int nc1 = ((col >> 1) << 2) | (col & 1);
    const int nc2 = nc1 + 2;
#pragma unroll
    for (int r = 0; r < 8; ++r) {
      M[r * 16 + nc1]        = sr[r];   //  Ar   (multiplies Re rows of V)
      M[(r + 8) * 16 + nc1]  = si[r];   //  Ai
      M[r * 16 + nc2]        = -si[r];  // -Ai   (multiplies Im rows of V)
      M[(r + 8) * 16 + nc2]  = sr[r];   //  Ar
    }
  }
}

// ---------------------------------------------------------------------------
// Feature layer: RY(pi t), RY(2 pi t), RY(3 pi t) on wires 0,1,2.
// Real rotation, applied identically to the 8 per-lane values (Re or Im half).
// ---------------------------------------------------------------------------
__device__ __forceinline__ void qb_feature(float* x, float tv) {
  const float h = 1.5707963267948966f * tv;  // 0.5 * pi * t
  float s0, c0, s1, c1, s2, c2;
  __sincosf(h, &s0, &c0);
  __sincosf(2.0f * h, &s1, &c1);
  __sincosf(3.0f * h, &s2, &c2);
#pragma unroll
  for (int j = 0; j < 4; ++j) {           // wire0: pairs (j, j+4)
    const float a = x[j], b = x[j + 4];
    x[j]     = c0 * a - s0 * b;
    x[j + 4] = s0 * a + c0 * b;
  }
#pragma unroll
  for (int g = 0; g < 2; ++g) {           // wire1: pairs (p, p+2)
#pragma unroll
    for (int j = 0; j < 2; ++j) {
      const int p = g * 4 + j;
      const float a = x[p], b = x[p + 2];
      x[p]     = c1 * a - s1 * b;
      x[p + 2] = s1 * a + c1 * b;
    }
  }
#pragma unroll
  for (int p = 0; p < 8; p += 2) {        // wire2: pairs (p, p+1)
    const float a = x[p], b = x[p + 1];
    x[p]     = c2 * a - s2 * b;
    x[p + 1] = s2 * a + c2 * b;
  }
}

// ---------------------------------------------------------------------------
// 16x16 (K-permuted) mat-vec over 16 batch columns: 4 chained f32 WMMAs.
// Thanks to the K-permutation, B chunk k is just {x[2k], x[2k+1]} on all lanes.
// ---------------------------------------------------------------------------
__device__ __forceinline__ void qb_matvec(float* x, const v2f* A) {
  v8f acc = {};
  acc = __builtin_amdgcn_wmma_f32_16x16x4_f32(false, A[0], false, mk2(x[0], x[1]), (short)0, acc, false, false);
  acc = __builtin_amdgcn_wmma_f32_16x16x4_f32(false, A[1], false, mk2(x[2], x[3]), (short)0, acc, false, false);
  acc = __builtin_amdgcn_wmma_f32_16x16x4_f32(false, A[2], false, mk2(x[4], x[5]), (short)0, acc, false, false);
  acc = __builtin_amdgcn_wmma_f32_16x16x4_f32(false, A[3], false, mk2(x[6], x[7]), (short)0, acc, false, false);
#pragma unroll
  for (int j = 0; j < 8; ++j) x[j] = acc[j];
}

__device__ __forceinline__ float qb_tile(const float* __restrict__ t, const float* v0,
                                         const v2f* A2, const v2f* A3, int tile, int lo16) {
  const float tv = t[tile * 16 + lo16];
  float x[8];
#pragma unroll
  for (int j = 0; j < 8; ++j) x[j] = v0[j];
  qb_feature(x, tv);
  qb_matvec(x, A2);
  qb_feature(x, tv);
  qb_matvec(x, A3);
  // <Z0> partial: rows 0-3 (+), rows 4-7 (-); same signs for Re and Im halves
  return x[0] * x[0] + x[1] * x[1] + x[2] * x[2] + x[3] * x[3]
       - x[4] * x[4] - x[5] * x[5] - x[6] * x[6] - x[7] * x[7];
}

__global__ __launch_bounds__(256) void qb_main(const float* __restrict__ t,
                                               const float* __restrict__ ws,
                                               float* __restrict__ out,
                                               int ntiles) {
  const int lane = threadIdx.x & 31;
  const int lo16 = lane & 15;
  const int hb = lane >> 4;  // 0 = Re half, 1 = Im half
  const int waveId = (blockIdx.x * blockDim.x + threadIdx.x) >> 5;

  // A-fragments from the (K-permuted) M: VGPR0 = K=(4k+2hb), VGPR1 = K+1; row = lane&15
  const float* M2 = ws + 16;
  const float* M3 = ws + 16 + 256;
  v2f A2[4], A3[4];
#pragma unroll
  for (int k = 0; k < 4; ++k) {
    const int c0 = 4 * k + 2 * hb;
    A2[k] = mk2(M2[lo16 * 16 + c0], M2[lo16 * 16 + c0 + 1]);
    A3[k] = mk2(M3[lo16 * 16 + c0], M3[lo16 * 16 + c0 + 1]);
  }
  float v0[8];
#pragma unroll
  for (int j = 0; j < 8; ++j) v0[j] = ws[hb * 8 + j];

#pragma unroll 1
  for (int it = 0; it < TILES_PER_WAVE; it += 2) {
    const int tile0 = waveId * TILES_PER_WAVE + it;
    if (tile0 >= ntiles) return;            // wave-uniform
    const int tile1 = tile0 + 1;
    const bool has1 = (tile1 < ntiles);     // wave-uniform

    // Two independent tiles: interleaved WMMA chains hide accumulation latency
    float p0 = qb_tile(t, v0, A2, A3, tile0, lo16);
    float p1 = has1 ? qb_tile(t, v0, A2, A3, tile1, lo16) : 0.0f;

    p0 += __shfl_xor(p0, 16, 32);           // combine Re^2 and Im^2 halves
    p1 += __shfl_xor(p1, 16, 32);
    if (hb == 0) {
      out[tile0 * 16 + lo16] = p0;
      if (has1) out[tile1 * 16 + lo16] = p1;
    }
  }
}

extern "C" void kernel_launch(void* const* d_in, const int* in_sizes, int n_in,
                              void* d_out, int out_size, void* d_ws, size_t ws_size,
                              hipStream_t stream) {
  (void)in_sizes; (void)n_in; (void)ws_size;
  const float* t = (const float*)d_in[0];       // (B,1) float32
  const float* theta = (const float*)d_in[1];   // (3,3,3) float32
  float* out = (float*)d_out;                   // (B,1) float32
  float* ws = (float*)d_ws;                     // needs 528 floats

  qb_setup<<<1, 32, 0, stream>>>(theta, ws);

  const int ntiles = out_size / 16;  // B / 16 samples per WMMA tile
  const int waves = (ntiles + TILES_PER_WAVE - 1) / TILES_PER_WAVE;
  const int threads = 256;  // 8 waves/block on wave32
  const int wavesPerBlock = threads / 32;
  const int blocks = (waves + wavesPerBlock - 1) / wavesPerBlock;
  qb_main<<<blocks, threads, 0, stream>>>(t, ws, out, ntiles);
}